// GCNLayer_72284299592299
// MI455X (gfx1250) — compile-verified
//
#include <hip/hip_runtime.h>

typedef float v2f __attribute__((ext_vector_type(2)));
typedef float v8f __attribute__((ext_vector_type(8)));

#define FEAT 16

// ---------------------------------------------------------------------------
// Guaranteed single hardware FP32 atomic add (no return -> STOREcnt tracked).
// Device scope so the RMW is performed coherently at the global L2.
// ---------------------------------------------------------------------------
__device__ __forceinline__ void atomic_add_f32_dev(float* p, float v) {
    asm volatile("global_atomic_add_f32 %0, %1, off scope:SCOPE_DEV"
                 :
                 : "v"(p), "v"(v)
                 : "memory");
}

// ---------------------------------------------------------------------------
// Kernel 1: zero the output accumulator and the two degree arrays.
// (harness poisons d_out/d_ws with 0xAA; we must re-zero every launch)
// ---------------------------------------------------------------------------
__global__ void gcn_zero_kernel(float* __restrict__ out, int out_n,
                                float* __restrict__ deg, int deg_n) {
    int i = blockIdx.x * blockDim.x + threadIdx.x;
    if (i < out_n) out[i] = 0.0f;
    if (i < deg_n) deg[i] = 0.0f;
}

// ---------------------------------------------------------------------------
// Kernel 2: degree counting. One thread per edge; two f32 atomics to L2.
// ---------------------------------------------------------------------------
__global__ void gcn_degree_kernel(const int* __restrict__ src,
                                  const int* __restrict__ dst,
                                  int E,
                                  float* __restrict__ out_deg,
                                  float* __restrict__ in_deg) {
    int e = blockIdx.x * blockDim.x + threadIdx.x;
    if (e >= E) return;
    int s = src[e];
    int d = dst[e];
    atomic_add_f32_dev(&out_deg[s], 1.0f);
    atomic_add_f32_dev(&in_deg[d], 1.0f);
}

// ---------------------------------------------------------------------------
// Kernel 3: deg -> 1/sqrt(max(deg,1)) in place (covers both arrays at once).
// ---------------------------------------------------------------------------
__global__ void gcn_scale_kernel(float* __restrict__ deg, int n) {
    int i = blockIdx.x * blockDim.x + threadIdx.x;
    if (i >= n) return;
    deg[i] = 1.0f / sqrtf(fmaxf(deg[i], 1.0f));
}

// ---------------------------------------------------------------------------
// Kernel 4: edge aggregation (hot pass: 80M f32 atomics, all L2-resident).
// Thread t -> (edge e = t/4, quad q = t%4). Each lane does ONE b128 gather
// (16B of the 64B feature row, 16B-aligned), scales by out_scale[src], and
// issues 4 consecutive global_atomic_add_f32 into out[dst].
// Per edge: 16 load-requests + 16 atomic-requests (vs 64+16 for scalar gathers).
// ---------------------------------------------------------------------------
__global__ void gcn_edge_kernel(const float4* __restrict__ feat4,
                                const float* __restrict__ oscale,
                                const int* __restrict__ src,
                                const int* __restrict__ dst,
                                int E4,
                                float* __restrict__ out) {
    int t = blockIdx.x * blockDim.x + threadIdx.x;
    if (t >= E4) return;
    int e = t >> 2;
    int q = t & 3;
    int s = src[e];
    int d = dst[e];
    float sc = oscale[s];
    float4 v = feat4[s * 4 + q];            // global_load_b128
    float* o = out + d * FEAT + q * 4;
    atomic_add_f32_dev(o + 0, v.x * sc);
    atomic_add_f32_dev(o + 1, v.y * sc);
    atomic_add_f32_dev(o + 2, v.z * sc);
    atomic_add_f32_dev(o + 3, v.w * sc);
}

// ---------------------------------------------------------------------------
// Kernel 5: post-normalization out = diag(in_scale) * out, done as a true
// 16x16x16 matmul per node tile via 4 chained V_WMMA_F32_16X16X4_F32.
//
// A_k (16x4): one-hot diagonal chunk, A_k[m][c] = s[m] iff m == 4k+c.
//   f32 A layout (wave32): lanes0-15 -> M=lane, v0:K=0 v1:K=1;
//                          lanes16-31 -> M=lane-16, v0:K=2 v1:K=3.
// B_k (4x16): rows 4k..4k+3 of the tile, N striped across lanes:
//   v0: lanes0-15 row 4k+0 / lanes16-31 row 4k+2; v1: rows 4k+1 / 4k+3.
// C/D (16x16 f32, 8 VGPRs): c[v] at lane L = element (M = v + 8*(L>=16), N=L&15).
//
// Each dot product has exactly one nonzero term, so D[m][n] = s[m]*X[m][n]
// exactly in f32. One wave owns one 16-node tile -> in-place RMW is race-free.
// Early-out is per-wave uniform, so EXEC is all-ones for every executing WMMA.
// ---------------------------------------------------------------------------
__global__ void gcn_postscale_wmma_kernel(float* __restrict__ out,
                                          const float* __restrict__ iscale,
                                          int ntiles) {
    int lane = threadIdx.x & 31;
    int tile = blockIdx.x * (blockDim.x >> 5) + (threadIdx.x >> 5);
    if (tile >= ntiles) return;

    int hi = lane >> 4;   // lane half
    int ln = lane & 15;   // N for B/D, M for A

    float s = iscale[tile * FEAT + ln];
    float* base = out + (size_t)tile * (FEAT * FEAT);

    v8f c = {};
#pragma unroll
    for (int k = 0; k < 4; ++k) {
        int r0 = 4 * k + 2 * hi;     // K index carried by v0 in this lane half
        v2f a, b;
        a.x = (ln == r0)     ? s : 0.0f;
        a.y = (ln == r0 + 1) ? s : 0.0f;
        b.x = base[r0 * FEAT + ln];
        b.y = base[(r0 + 1) * FEAT + ln];
        c = __builtin_amdgcn_wmma_f32_16x16x4_f32(
                /*neg_a=*/false, a, /*neg_b=*/false, b,
                /*c_mod=*/(short)0, c, /*reuse_a=*/false, /*reuse_b=*/false);
    }

#pragma unroll
    for (int v = 0; v < 8; ++v) {
        base[(v + 8 * hi) * FEAT + ln] = c[v];
    }
}

// Scalar tail for node counts not divisible by 16 (not hit for N=100000).
__global__ void gcn_postscale_tail_kernel(float* __restrict__ out,
                                          const float* __restrict__ iscale,
                                          int start_node, int n) {
    int i = start_node * FEAT + blockIdx.x * blockDim.x + threadIdx.x;
    if (i >= n * FEAT) return;
    out[i] *= iscale[i / FEAT];
}

extern "C" void kernel_launch(void* const* d_in, const int* in_sizes, int n_in,
                              void* d_out, int out_size, void* d_ws, size_t ws_size,
                              hipStream_t stream) {
    const float*  feat  = (const float*)d_in[0];
    const float4* feat4 = (const float4*)d_in[0];
    const int*    src   = (const int*)d_in[1];   // int32 per harness convention
    const int*    dst   = (const int*)d_in[2];
    float*        out   = (float*)d_out;

    const int feat_total = in_sizes[0];          // N * 16
    const int n          = feat_total / FEAT;    // 100000
    const int E          = in_sizes[1];          // 5000000
    (void)feat;

    float* out_deg = (float*)d_ws;               // n floats -> becomes out_scale
    float* in_deg  = out_deg + n;                // n floats -> becomes in_scale

    const int B = 256;

    // 1) zero accumulator + degrees
    {
        int zn = feat_total > 2 * n ? feat_total : 2 * n;
        gcn_zero_kernel<<<(zn + B - 1) / B, B, 0, stream>>>(out, feat_total, out_deg, 2 * n);
    }

    // 2) degrees
    gcn_degree_kernel<<<(E + B - 1) / B, B, 0, stream>>>(src, dst, E, out_deg, in_deg);

    // 3) deg -> rsqrt(max(deg,1)) for both arrays
    gcn_scale_kernel<<<(2 * n + B - 1) / B, B, 0, stream>>>(out_deg, 2 * n);

    // 4) edge aggregation (dominant pass): 4 lanes per edge, b128 gathers
    {
        int e4 = E * 4;  // 20,000,000
        gcn_edge_kernel<<<(e4 + B - 1) / B, B, 0, stream>>>(feat4, out_deg, src, dst, e4, out);
    }

    // 5) post-normalize with WMMA (one wave per 16-node tile)
    {
        int ntiles = n / FEAT;
        int waves_per_block = B / 32;
        if (ntiles > 0) {
            gcn_postscale_wmma_kernel<<<(ntiles + waves_per_block - 1) / waves_per_block,
                                        B, 0, stream>>>(out, in_deg, ntiles);
        }
        int rem_start = ntiles * FEAT;
        if (rem_start < n) {
            int rem_elems = (n - rem_start) * FEAT;
            gcn_postscale_tail_kernel<<<(rem_elems + B - 1) / B, B, 0, stream>>>(
                out, in_deg, rem_start, n);
        }
    }
}